// Hodge1LapEdgeEncoder_12232066859390
// MI455X (gfx1250) — compile-verified
//
#include <hip/hip_runtime.h>
#include <hip/hip_bf16.h>

// CDNA5 / gfx1250 (wave32). D = A*B + C WMMA, f32 16x16x4.
typedef __attribute__((ext_vector_type(2))) float v2f;
typedef __attribute__((ext_vector_type(8))) float v8f;

// ---------------------------------------------------------------------------
// Kernel 1 (fused): initialize out_val with the broadcast padding row
// (B128 stores) AND write the full-clique block-diagonal index.
// Grid covers total4 = P*D/4 >= P, so the first P threads also emit idx:
//   p = g*n*n + i*n + j -> rows[p] = p / n ; cols[p] = (p/(n*n))*n + p%n
// Output (64 MB) fits in the 192 MB L2 with RT hints, so the scatter
// kernel's atomics afterwards resolve in L2.
// ---------------------------------------------------------------------------
__global__ void hodge_fill_idx_kernel(const float4* __restrict__ padding4,
                                      float4* __restrict__ out_val4,
                                      int* __restrict__ out_idx,
                                      int total4, int dquads, int dqmask,
                                      int P, int n) {
    int i = blockIdx.x * blockDim.x + threadIdx.x;
    if (i >= total4) return;
    int q = (dqmask >= 0) ? (i & dqmask) : (i % dquads);  // uniform branch
    out_val4[i] = padding4[q];
    if (i < P) {
        int nn   = n * n;
        int rows = i / n;
        int cols = (i / nn) * n + (i % n);
        out_idx[i]     = rows;
        out_idx[P + i] = cols;
    }
}

// ---------------------------------------------------------------------------
// Kernel 2 (fast path, D compile-time, every wave owns exactly 16 valid
// edges): scatter-add  val[e,d] = edge_attr[e,d] + pestat[e,0]*fc_w[d]
// into out_val at pair (u*n + v%n), via V_WMMA_F32_16X16X4_F32.
//
//   A (16x4 f32): lanes 0-15 K=0 slot <- pestat[e0+lane,0]; K=1..3 zero.
//   B (4x16  f32): lanes 0-15 K=0 row <- fc_w[d0+lane];     K=1..3 zero.
//   C (16x16 f32): VGPR r, lane L -> (M = r + 8*(L>=16), N = L&15),
//     preloaded with the edge_attr tile, so D = p (x) w + edge_attr exactly.
//
// Branch-free: out-of-role lanes use clamped in-bounds loads + v_cndmask
// zeroing, so EXEC stays all-1s (WMMA requirement). edge_attr loads use one
// base address + immediate offsets; atomics use 8 hoisted base pointers +
// immediate offsets.
// ---------------------------------------------------------------------------
template <int DT>
__global__ void hodge_scatter_fast(const int*   __restrict__ edge_index,
                                   const float* __restrict__ pestat,
                                   const float* __restrict__ edge_attr,
                                   const float* __restrict__ fc_w,
                                   float*       __restrict__ out_val,
                                   int E, int Kpe, int n) {
    const int lane = threadIdx.x & 31;
    const int wave = threadIdx.x >> 5;
    const int wavesPerBlock = blockDim.x >> 5;
    const int e0 = (blockIdx.x * wavesPerBlock + wave) * 16;

    const int half = lane >> 4;               // 0: lanes 0-15, 1: lanes 16-31
    const int l16  = lane & 15;

    // All lanes load (clamped in-bounds); role-zeroing is a cndmask.
    int   ec = min(e0 + lane, E - 1);
    float pv = pestat[(size_t)ec * Kpe];                  // PE column 0
    int   u  = edge_index[ec];                            // g*n + lu
    int   v  = edge_index[E + ec];                        // g*n + lv
    int   pairBase = (u * n + (v % n)) * DT;              // lanes>=16: garbage, never read

    v2f a;
    a.x = (lane < 16) ? pv : 0.0f;            // A: only K=0 slot nonzero
    a.y = 0.0f;

    // Element r of this lane's accumulator belongs to edge M = r + 8*half,
    // whose pair base lives in lane M (< 16). Hoist 8 atomic base pointers.
    float* op[8];
#pragma unroll
    for (int r = 0; r < 8; ++r) {
        int pb = __shfl(pairBase, r + half * 8, 32);
        op[r] = out_val + pb + l16;
    }

    // One base address; all 32 loads use immediate offsets (r*DT + d0 < 2KB).
    const float* abase = edge_attr + (size_t)(e0 + half * 8) * DT + l16;

#pragma unroll
    for (int t = 0; t < DT / 16; ++t) {
        const int d0 = t * 16;
        float w = fc_w[d0 + l16];             // in-bounds for all lanes
        v2f b;
        b.x = (lane < 16) ? w : 0.0f;         // B: only K=0 row nonzero
        b.y = 0.0f;

        v8f c;
#pragma unroll
        for (int r = 0; r < 8; ++r)
            c[r] = abase[r * DT + d0];

        // emits v_wmma_f32_16x16x4_f32  (D = A*B + C, all f32)
        c = __builtin_amdgcn_wmma_f32_16x16x4_f32(
                /*neg_a=*/false, a, /*neg_b=*/false, b,
                /*c_mod=*/(short)0, c, /*reuse_a=*/false, /*reuse_b=*/false);

#pragma unroll
        for (int r = 0; r < 8; ++r)
            atomicAdd(op[r] + d0, c[r]);      // immediate offset d0*4
    }
}

// ---------------------------------------------------------------------------
// Kernel 2 (generic fallback): guarded version for shapes that don't match
// the fast path (D % 16 == 0 still required by the WMMA tiling).
// ---------------------------------------------------------------------------
__global__ void hodge_scatter_generic(const int*   __restrict__ edge_index,
                                      const float* __restrict__ pestat,
                                      const float* __restrict__ edge_attr,
                                      const float* __restrict__ fc_w,
                                      float*       __restrict__ out_val,
                                      int E, int Kpe, int D, int n) {
    const int lane = threadIdx.x & 31;
    const int wave = threadIdx.x >> 5;
    const int wavesPerBlock = blockDim.x >> 5;
    const int e0 = (blockIdx.x * wavesPerBlock + wave) * 16;
    if (e0 >= E) return;                      // wave-uniform exit

    const int half = lane >> 4;
    const int l16  = lane & 15;

    float p = 0.0f;
    int   pairBase = 0;
    if (lane < 16) {
        int e = e0 + lane;
        if (e < E) {
            p = pestat[(size_t)e * Kpe];
            int u = edge_index[e];
            int v = edge_index[E + e];
            pairBase = (u * n + (v % n)) * D;
        }
    }

    v2f a; a.x = p; a.y = 0.0f;

    int pb[8];
#pragma unroll
    for (int r = 0; r < 8; ++r)
        pb[r] = __shfl(pairBase, r + half * 8, 32);

    const int tiles = D >> 4;
    for (int t = 0; t < tiles; ++t) {
        const int d0 = t * 16;
        float w = (lane < 16) ? fc_w[d0 + l16] : 0.0f;
        v2f b; b.x = w; b.y = 0.0f;

        v8f c;
#pragma unroll
        for (int r = 0; r < 8; ++r) {
            int e = e0 + r + half * 8;
            c[r] = (e < E) ? edge_attr[(size_t)e * D + d0 + l16] : 0.0f;
        }

        c = __builtin_amdgcn_wmma_f32_16x16x4_f32(
                false, a, false, b, (short)0, c, false, false);

#pragma unroll
        for (int r = 0; r < 8; ++r) {
            int e = e0 + r + half * 8;
            if (e < E)
                atomicAdd(out_val + pb[r] + d0 + l16, c[r]);
        }
    }
}

// ---------------------------------------------------------------------------
// Host launcher. Inputs (setup_inputs order):
//   0 edge_index [2,E] i32 | 1 pestat [E,K] f32 | 2 edge_attr [E,D] f32
//   3 batch [N] i32        | 4 fc_w [D] f32     | 5 padding [D] f32
//   6 n_graphs i32         | 7 nodes_per_graph i32
// Output: out_idx (2*P int32) || out_val (P*D f32),  P = B*n*n.
// ---------------------------------------------------------------------------
extern "C" void kernel_launch(void* const* d_in, const int* in_sizes, int n_in,
                              void* d_out, int out_size, void* d_ws, size_t ws_size,
                              hipStream_t stream) {
    const int*   edge_index = (const int*)  d_in[0];
    const float* pestat     = (const float*)d_in[1];
    const float* edge_attr  = (const float*)d_in[2];
    const float* fc_w       = (const float*)d_in[4];
    const float* padding    = (const float*)d_in[5];

    const int D   = in_sizes[4];             // 64
    const int E   = in_sizes[2] / D;         // 65536
    const int Kpe = in_sizes[1] / E;         // 8
    const int N   = in_sizes[3];             // B*n = 2048
    const int P   = out_size / (D + 2);      // B*n*n = 262144
    const int n   = P / N;                   // 128

    int*   out_idx = (int*)d_out;
    float* out_val = (float*)d_out + (size_t)2 * P;

    // Kernel 1: fill out_val with padding + write indices (fused).
    {
        int total4  = (int)(((size_t)P * D) / 4);
        int dquads  = D / 4;
        int dqmask  = ((dquads & (dquads - 1)) == 0) ? (dquads - 1) : -1;
        int threads = 256;
        int blocks  = (total4 + threads - 1) / threads;
        hodge_fill_idx_kernel<<<blocks, threads, 0, stream>>>(
            (const float4*)padding, (float4*)out_val, out_idx,
            total4, dquads, dqmask, P, n);
    }

    // Kernel 2: WMMA-fused scatter-add (stream order: runs after the fill).
    {
        int threads       = 256;                       // 8 waves
        int edgesPerBlock = (threads / 32) * 16;       // 128
        int blocks        = (E + edgesPerBlock - 1) / edgesPerBlock;  // 512
        if (D == 64 && (E % edgesPerBlock) == 0) {
            hodge_scatter_fast<64><<<blocks, threads, 0, stream>>>(
                edge_index, pestat, edge_attr, fc_w, out_val, E, Kpe, n);
        } else {
            hodge_scatter_generic<<<blocks, threads, 0, stream>>>(
                edge_index, pestat, edge_attr, fc_w, out_val, E, Kpe, D, n);
        }
    }
}